// GINEClassifier_27118423507097
// MI455X (gfx1250) — compile-verified
//
#include <hip/hip_runtime.h>
#include <math.h>

// ---------------------------------------------------------------------------
// Problem constants (match reference)
// ---------------------------------------------------------------------------
#define N_NODES   50000
#define N_EDGES   1600000
#define N_GRAPHS  128
#define HID       64
#define EDGE_DIM  16
#define N_CLASSES 10
#define N_LAYERS  5
#define BN_EPS    1e-5f

typedef float v2f __attribute__((ext_vector_type(2)));
typedef float v8f __attribute__((ext_vector_type(8)));

// fp32 no-return atomic add (lowers to global_atomic_add_f32, SCOPE_DEV)
__device__ __forceinline__ void atomicAddF32(float* p, float v) {
    __hip_atomic_fetch_add(p, v, __ATOMIC_RELAXED, __HIP_MEMORY_SCOPE_AGENT);
}

// ---------------------------------------------------------------------------
// Persistent 64-wide GEMM:  OUT[r, :] = act( Afetch(r,:) @ W(64x64) + bias )
//   FUSE=1 : A = (1+eps)*A0 + AGG   (GINE combine), else A = A0
//   RELU=1 : relu on output
// Weights hoisted into 128 VGPRs per lane once per wave (grid-stride over row
// tiles). Inner loop: 16 contiguous A-pair loads (constant offsets off one
// base), 64 WMMAs, 32 stores (constant offsets off one base).
// WMMA f32 16x16x4; A/B K-index held by VGPR v is (4*kk + 2*half + v).
// ---------------------------------------------------------------------------
template <bool FUSE, bool RELU>
__global__ __launch_bounds__(128) void gemm64_wmma(
    const float* __restrict__ A0, const float* __restrict__ W,
    const float* __restrict__ bias, const float* __restrict__ AGG,
    const float* __restrict__ epsp, float* __restrict__ OUT, int ntiles)
{
    const int wave = threadIdx.x >> 5;
    const int lane = threadIdx.x & 31;
    const int half = lane >> 4;            // 0 or 1
    const int m    = lane & 15;            // row within tile / col within n-tile

    float s = 1.0f;
    if (FUSE) s = 1.0f + epsp[0];

    // hoist full B operand + bias into registers (same for every tile)
    v2f wreg[16][4];
    float bb[4];
    #pragma unroll
    for (int kk = 0; kk < 16; ++kk) {
        const int kb = kk * 4 + 2 * half;
        #pragma unroll
        for (int n = 0; n < 4; ++n) {
            wreg[kk][n].x = W[kb * 64       + n * 16 + m];
            wreg[kk][n].y = W[(kb + 1) * 64 + n * 16 + m];
        }
    }
    #pragma unroll
    for (int n = 0; n < 4; ++n) bb[n] = bias[n * 16 + m];

    const int wstride = gridDim.x * 4;
    for (int tile = blockIdx.x * 4 + wave; tile < ntiles; tile += wstride) {
        const int r0    = tile * 16;
        const int abase = (r0 + m) * 64 + 2 * half;   // + kk*4 (const) per step

        v8f acc[4];
        acc[0] = v8f{}; acc[1] = v8f{}; acc[2] = v8f{}; acc[3] = v8f{};

        #pragma unroll
        for (int kk = 0; kk < 16; ++kk) {
            v2f a;
            if (FUSE) {
                a.x = s * A0[abase + kk * 4]     + AGG[abase + kk * 4];
                a.y = s * A0[abase + kk * 4 + 1] + AGG[abase + kk * 4 + 1];
            } else {
                a.x = A0[abase + kk * 4];
                a.y = A0[abase + kk * 4 + 1];
            }
            #pragma unroll
            for (int n = 0; n < 4; ++n) {
                acc[n] = __builtin_amdgcn_wmma_f32_16x16x4_f32(
                    false, a, false, wreg[kk][n], (short)0, acc[n], false, false);
            }
        }

        // store: one base, 32 constant offsets
        float* op = OUT + (r0 + 8 * half) * 64 + m;
        #pragma unroll
        for (int n = 0; n < 4; ++n) {
            #pragma unroll
            for (int v = 0; v < 8; ++v) {
                float val = acc[n][v] + bb[n];
                if (RELU) val = fmaxf(val, 0.0f);
                op[v * 64 + n * 16] = val;
            }
        }
    }
}

// ---------------------------------------------------------------------------
// Fused edge kernel (persistent): for each edge
//     msg = relu(h[src] + (ea @ We + be));  aggr[dst] += msg
// e is recomputed with WMMA f32 16x16x4 (16-edge tile, K=16, N=64).
// Entire We (16x64) lives in 32 VGPRs per lane. Per tile: 16 index loads,
// 4 A-pair loads, 16 WMMAs, then 32 batched gathers (single wait) and
// 32 streamed atomics; per-element address ALU folded into constant
// instruction offsets (n*16 elements = 64*n bytes).
// C-layout: VGPR v of lane -> edge (v + 8*half), channel (n*16 + m).
// ---------------------------------------------------------------------------
__global__ __launch_bounds__(128) void edge_msg_aggr(
    const float* __restrict__ EA, const int* __restrict__ src,
    const int* __restrict__ dst, const float* __restrict__ We,
    const float* __restrict__ be, const float* __restrict__ H,
    float* __restrict__ AGGR, int ntiles)
{
    const int wave = threadIdx.x >> 5;
    const int lane = threadIdx.x & 31;
    const int half = lane >> 4;
    const int m    = lane & 15;

    // hoist full We + be into registers (K=16 -> 4 ksteps x 4 ntiles x v2f)
    v2f wreg[4][4];
    float bb[4];
    #pragma unroll
    for (int kk = 0; kk < 4; ++kk) {
        const int kb = kk * 4 + 2 * half;
        #pragma unroll
        for (int n = 0; n < 4; ++n) {
            wreg[kk][n].x = We[kb * 64       + n * 16 + m];
            wreg[kk][n].y = We[(kb + 1) * 64 + n * 16 + m];
        }
    }
    #pragma unroll
    for (int n = 0; n < 4; ++n) bb[n] = be[n * 16 + m];

    const int wstride = gridDim.x * 4;
    for (int tile = blockIdx.x * 4 + wave; tile < ntiles; tile += wstride) {
        const int e0 = tile * 16;

        // gather/scatter element offsets for this lane's 8 C rows
        int so[8], dO[8];
        #pragma unroll
        for (int v = 0; v < 8; ++v) {
            const int e = e0 + v + 8 * half;
            so[v] = src[e] * 64 + m;   // + n*16 folds into inst offset
            dO[v] = dst[e] * 64 + m;
        }

        v8f acc[4];
        acc[0] = v8f{}; acc[1] = v8f{}; acc[2] = v8f{}; acc[3] = v8f{};

        const int abase = (e0 + m) * EDGE_DIM + 2 * half;
        #pragma unroll
        for (int kk = 0; kk < 4; ++kk) {
            v2f a;
            a.x = EA[abase + kk * 4];
            a.y = EA[abase + kk * 4 + 1];
            #pragma unroll
            for (int n = 0; n < 4; ++n) {
                acc[n] = __builtin_amdgcn_wmma_f32_16x16x4_f32(
                    false, a, false, wreg[kk][n], (short)0, acc[n], false, false);
            }
        }

        // batch all 32 gathers first (clause + single wait) ...
        float hv[4][8];
        #pragma unroll
        for (int n = 0; n < 4; ++n) {
            #pragma unroll
            for (int v = 0; v < 8; ++v) hv[n][v] = H[so[v] + n * 16];
        }
        // ... then stream 32 relu+atomics
        #pragma unroll
        for (int n = 0; n < 4; ++n) {
            #pragma unroll
            for (int v = 0; v < 8; ++v) {
                float msg = fmaxf(acc[n][v] + bb[n] + hv[n][v], 0.0f);
                atomicAddF32(&AGGR[dO[v] + n * 16], msg);
            }
        }
    }
}

// ---------------------------------------------------------------------------
// BatchNorm stats: per-channel sum & sum-of-squares into stats[0:64], [64:128]
// ---------------------------------------------------------------------------
__global__ void bn_stats(const float* __restrict__ Z, float* __restrict__ stats,
                         int total)
{
    const int idx = blockIdx.x * blockDim.x + threadIdx.x;
    const int T   = gridDim.x * blockDim.x;     // multiple of 64
    const int c   = idx & 63;
    float s = 0.0f, s2 = 0.0f;
    for (int i = idx; i < total; i += T) {
        const float v = Z[i];
        s  += v;
        s2 += v * v;
    }
    atomicAddF32(&stats[c], s);
    atomicAddF32(&stats[64 + c], s2);
}

// h = relu(bn(z)) + h_res   (h buffer still holds h_res)
__global__ void bn_apply(const float* __restrict__ Z,
                         const float* __restrict__ stats,
                         const float* __restrict__ gamma,
                         const float* __restrict__ beta,
                         float* __restrict__ H, int total)
{
    const int i = blockIdx.x * blockDim.x + threadIdx.x;
    if (i >= total) return;
    const int   c    = i & 63;
    const float invN = 1.0f / (float)N_NODES;
    const float mu   = stats[c] * invN;
    const float var  = stats[64 + c] * invN - mu * mu;
    const float rs   = rsqrtf(var + BN_EPS);
    const float z    = gamma[c] * (Z[i] - mu) * rs + beta[c];
    H[i] = fmaxf(z, 0.0f) + H[i];
}

// ---------------------------------------------------------------------------
// Global mean pool (accumulate)
// ---------------------------------------------------------------------------
__global__ void pool_accum(const float* __restrict__ H,
                           const int* __restrict__ batch,
                           float* __restrict__ gsum, float* __restrict__ cnt,
                           int total)
{
    const int i = blockIdx.x * blockDim.x + threadIdx.x;
    if (i >= total) return;
    const int n = i >> 6;
    const int c = i & 63;
    const int b = batch[n];
    atomicAddF32(&gsum[b * 64 + c], H[i]);
    if (c == 0) atomicAddF32(&cnt[b], 1.0f);
}

// ---------------------------------------------------------------------------
// Classifier head: one block (64 threads) per graph
// out layout: logits[1280] | probs[1280] | preds[1280] | preds[1280]
// ---------------------------------------------------------------------------
__global__ __launch_bounds__(64) void classifier(
    const float* __restrict__ gsum, const float* __restrict__ cnt,
    const float* __restrict__ W1, const float* __restrict__ b1,
    const float* __restrict__ W2, const float* __restrict__ b2,
    float* __restrict__ out)
{
    __shared__ float gl[64];
    __shared__ float al[64];
    const int g = blockIdx.x;
    const int c = threadIdx.x;

    const float denom = fmaxf(cnt[g], 1.0f);
    gl[c] = gsum[g * 64 + c] / denom;
    __syncthreads();

    float a = b1[c];
    #pragma unroll
    for (int k = 0; k < 64; ++k) a += gl[k] * W1[k * 64 + c];
    al[c] = fmaxf(a, 0.0f);
    __syncthreads();

    if (c < N_CLASSES) {
        float logit = b2[c];
        #pragma unroll
        for (int k = 0; k < 64; ++k) logit += al[k] * W2[k * N_CLASSES + c];
        const float prob = 1.0f / (1.0f + __expf(-logit));
        const float pred = prob > 0.5f ? 1.0f : 0.0f;
        const int o = g * N_CLASSES + c;
        out[o]                              = logit;
        out[N_GRAPHS * N_CLASSES     + o]   = prob;
        out[2 * N_GRAPHS * N_CLASSES + o]   = pred;
        out[3 * N_GRAPHS * N_CLASSES + o]   = pred;
    }
}

__global__ void zero_f32(float* __restrict__ p, int n)
{
    const int i = blockIdx.x * blockDim.x + threadIdx.x;
    if (i < n) p[i] = 0.0f;
}

// ---------------------------------------------------------------------------
// Host-side orchestration
// ---------------------------------------------------------------------------
extern "C" void kernel_launch(void* const* d_in, const int* in_sizes, int n_in,
                              void* d_out, int out_size, void* d_ws, size_t ws_size,
                              hipStream_t stream)
{
    (void)in_sizes; (void)n_in; (void)out_size; (void)ws_size;

    const float* x     = (const float*)d_in[0];
    const int*   ei    = (const int*)d_in[1];
    const int*   batch = (const int*)d_in[2];
    const float* ea    = (const float*)d_in[3];
    const float* Wn    = (const float*)d_in[4];
    const float* bnb   = (const float*)d_in[5];
    const float* We    = (const float*)d_in[6];
    const float* be    = (const float*)d_in[7];
    const float* eps   = (const float*)d_in[8];
    const float* W1    = (const float*)d_in[9];
    const float* b1    = (const float*)d_in[10];
    const float* W2    = (const float*)d_in[11];
    const float* b2    = (const float*)d_in[12];
    const float* gam   = (const float*)d_in[13];
    const float* bet   = (const float*)d_in[14];
    const float* cW1   = (const float*)d_in[15];
    const float* cb1   = (const float*)d_in[16];
    const float* cW2   = (const float*)d_in[17];
    const float* cb2   = (const float*)d_in[18];
    float*       out   = (float*)d_out;

    const int* src = ei;
    const int* dst = ei + N_EDGES;

    // workspace layout (floats)
    const int NH = N_NODES * HID;              // 3,200,000
    float* h     = (float*)d_ws;
    float* t     = h + NH;
    float* z     = t + NH;
    float* aggr  = z + NH;
    float* stats = aggr + NH;                  // 128
    float* gsum  = stats + 128;                // 8192
    float* cnt   = gsum + N_GRAPHS * HID;      // 128

    const int node_tiles  = N_NODES / 16;      // 3125 (exact)
    const int edge_tiles  = N_EDGES / 16;      // 100000 (exact)
    const int gemm_blocks = 256;               // persistent: 1024 waves
    const int edge_blocks = 2048;              // persistent: 8192 waves
    const int elem_blocks = (NH + 255) / 256;

    // h = x @ Wn + bn_
    gemm64_wmma<false, false><<<gemm_blocks, 128, 0, stream>>>(
        x, Wn, bnb, nullptr, nullptr, h, node_tiles);

    for (int l = 0; l < N_LAYERS; ++l) {
        // zero aggr + stats (contiguous)
        zero_f32<<<(NH + 128 + 255) / 256, 256, 0, stream>>>(aggr, NH + 128);

        // msg = relu(h[src] + (ea@We + be)); aggr[dst] += msg
        edge_msg_aggr<<<edge_blocks, 128, 0, stream>>>(
            ea, src, dst, We, be, h, aggr, edge_tiles);

        // t = relu( ((1+eps)*h + aggr) @ W1 + b1 )
        gemm64_wmma<true, true><<<gemm_blocks, 128, 0, stream>>>(
            h, W1 + l * HID * HID, b1 + l * HID, aggr, eps + l, t, node_tiles);

        // z = t @ W2 + b2
        gemm64_wmma<false, false><<<gemm_blocks, 128, 0, stream>>>(
            t, W2 + l * HID * HID, b2 + l * HID, nullptr, nullptr, z, node_tiles);

        // batch-norm (training-mode batch stats) + relu + residual
        bn_stats<<<512, 256, 0, stream>>>(z, stats, NH);
        bn_apply<<<elem_blocks, 256, 0, stream>>>(
            z, stats, gam + l * HID, bet + l * HID, h, NH);
    }

    // global mean pool
    zero_f32<<<(N_GRAPHS * HID + N_GRAPHS + 255) / 256, 256, 0, stream>>>(
        gsum, N_GRAPHS * HID + N_GRAPHS);
    pool_accum<<<elem_blocks, 256, 0, stream>>>(h, batch, gsum, cnt, NH);

    // classifier head -> logits | probs | preds | preds
    classifier<<<N_GRAPHS, 64, 0, stream>>>(gsum, cnt, cW1, cb1, cW2, cb2, out);
}